// GCNConv_27908697490045
// MI455X (gfx1250) — compile-verified
//
#include <hip/hip_runtime.h>

#define N_NODES 50000
#define N_EDGES 800000
#define D 128
#define M_TILES ((N_NODES + 15) / 16)   // 3125, exact

typedef __attribute__((ext_vector_type(2))) float v2f;
typedef __attribute__((ext_vector_type(8))) float v8f;

// ---------------------------------------------------------------------------
// 1) degree accumulation: deg[dst] += 1  (fp32 hardware atomics, L2-resident)
// ---------------------------------------------------------------------------
__global__ void GCN_deg_kernel(const int* __restrict__ edge_dst,
                               float* __restrict__ deg) {
    int e = blockIdx.x * blockDim.x + threadIdx.x;
    if (e < N_EDGES) {
        unsafeAtomicAdd(&deg[edge_dst[e]], 1.0f);
    }
}

// ---------------------------------------------------------------------------
// 2) inv_sqrt_deg[i] = rsqrt(deg[i] + 1)
// ---------------------------------------------------------------------------
__global__ void GCN_rsqrt_kernel(const float* __restrict__ deg,
                                 float* __restrict__ isd) {
    int i = blockIdx.x * blockDim.x + threadIdx.x;
    if (i < N_NODES) {
        isd[i] = rsqrtf(deg[i] + 1.0f);
    }
}

// ---------------------------------------------------------------------------
// 3) XW = X @ W via V_WMMA_F32_16X16X4_F32 (exact fp32 path).
//    One wave per 16-row M-tile; wave computes the full 16x128 output strip
//    (8 N-tiles, 64 accumulator VGPRs), reusing the A fragment across the 8
//    WMMAs of each k-step.
//
//    A 16x4 f32 layout : lanes 0-15 -> M=lane, VGPR0=K0,VGPR1=K1
//                        lanes 16-31 -> M=lane-16, VGPR0=K2,VGPR1=K3
//    B 4x16 f32 layout : VGPR0: lanes0-15 row K0, lanes16-31 row K2
//                        VGPR1: lanes0-15 row K1, lanes16-31 row K3
//    C/D 16x16 f32     : VGPR r: lanes0-15 M=r, lanes16-31 M=r+8; N=lane%16
// ---------------------------------------------------------------------------
__global__ __launch_bounds__(256) void GCN_gemm_kernel(
        const float* __restrict__ X, const float* __restrict__ W,
        float* __restrict__ XW) {
    const int lane = threadIdx.x & 31;
    const int wave = threadIdx.x >> 5;
    const int mt   = blockIdx.x * 8 + wave;          // 16-row tile index
    if (mt >= M_TILES) return;
    const int m0   = mt * 16;
    const int half = lane >> 4;                      // 0: lanes 0-15, 1: 16-31
    const int l16  = lane & 15;

    v8f c[8];
    const v8f zero = {};
    #pragma unroll
    for (int n = 0; n < 8; ++n) c[n] = zero;

    const float* xrow = X + (size_t)(m0 + l16) * D;  // this lane's A row

    for (int k0 = 0; k0 < D; k0 += 4) {
        v2f a;
        a.x = xrow[k0 + 2 * half + 0];               // K = k0 + {0 | 2}
        a.y = xrow[k0 + 2 * half + 1];               // K = k0 + {1 | 3}
        #pragma unroll
        for (int n = 0; n < 8; ++n) {
            const float* wr = W + (size_t)(k0 + 2 * half) * D + n * 16 + l16;
            v2f b;
            b.x = wr[0];                             // row K = k0 + 2*half
            b.y = wr[D];                             // row K = k0 + 2*half + 1
            c[n] = __builtin_amdgcn_wmma_f32_16x16x4_f32(
                /*neg_a=*/false, a, /*neg_b=*/false, b,
                /*c_mod=*/(short)0, c[n], /*reuse_a=*/false, /*reuse_b=*/false);
        }
    }

    #pragma unroll
    for (int n = 0; n < 8; ++n) {
        #pragma unroll
        for (int r = 0; r < 8; ++r) {
            XW[(size_t)(m0 + r + 8 * half) * D + n * 16 + l16] = c[n][r];
        }
    }
}

// ---------------------------------------------------------------------------
// 4) out[i] = XW[i] * (isd[i]^2)   (self-loop term; fully overwrites d_out)
//    One thread per (node, 4 columns) -> float4 path.
// ---------------------------------------------------------------------------
__global__ void GCN_selfloop_kernel(const float* __restrict__ XW,
                                    const float* __restrict__ isd,
                                    float* __restrict__ out) {
    int gid = blockIdx.x * blockDim.x + threadIdx.x;
    int i = gid >> 5;            // node
    int q = gid & 31;            // which float4 of the 128-wide row
    if (i >= N_NODES) return;
    float s = isd[i];
    s = s * s;                   // 1/deg
    float4 v = ((const float4*)(XW + (size_t)i * D))[q];
    float4 r = make_float4(v.x * s, v.y * s, v.z * s, v.w * s);
    ((float4*)(out + (size_t)i * D))[q] = r;
}

// ---------------------------------------------------------------------------
// 5) edge scatter: out[dst] += XW[src] * (isd[src]*isd[dst])
//    One wave per edge; lane l owns columns 4l..4l+3 (512B coalesced gather,
//    4 global_atomic_add_f32 per lane; out fits in 192MB L2 -> RMW stays hot).
// ---------------------------------------------------------------------------
__global__ void GCN_scatter_kernel(const int* __restrict__ edge_src,
                                   const int* __restrict__ edge_dst,
                                   const float* __restrict__ XW,
                                   const float* __restrict__ isd,
                                   float* __restrict__ out) {
    int gid = blockIdx.x * blockDim.x + threadIdx.x;
    int e = gid >> 5;
    int q = gid & 31;
    if (e >= N_EDGES) return;
    int s = edge_src[e];
    int d = edge_dst[e];
    float norm = isd[s] * isd[d];
    float4 v = ((const float4*)(XW + (size_t)s * D))[q];
    float* o = out + (size_t)d * D + q * 4;
    unsafeAtomicAdd(o + 0, v.x * norm);
    unsafeAtomicAdd(o + 1, v.y * norm);
    unsafeAtomicAdd(o + 2, v.z * norm);
    unsafeAtomicAdd(o + 3, v.w * norm);
}

// ---------------------------------------------------------------------------
extern "C" void kernel_launch(void* const* d_in, const int* in_sizes, int n_in,
                              void* d_out, int out_size, void* d_ws, size_t ws_size,
                              hipStream_t stream) {
    const float* X    = (const float*)d_in[0];
    const float* W    = (const float*)d_in[1];
    const int*   esrc = (const int*)d_in[2];
    const int*   edst = (const int*)d_in[3];
    float* out = (float*)d_out;

    // workspace layout: deg[N] | isd[N] | XW[N*D]   (~26 MB)
    float* deg = (float*)d_ws;
    float* isd = deg + N_NODES;
    float* XW  = deg + 2 * N_NODES;

    hipMemsetAsync(deg, 0, N_NODES * sizeof(float), stream);

    GCN_deg_kernel<<<(N_EDGES + 255) / 256, 256, 0, stream>>>(edst, deg);
    GCN_rsqrt_kernel<<<(N_NODES + 255) / 256, 256, 0, stream>>>(deg, isd);
    GCN_gemm_kernel<<<(M_TILES + 7) / 8, 256, 0, stream>>>(X, W, XW);

    const int self_threads = N_NODES * 32;
    GCN_selfloop_kernel<<<(self_threads + 255) / 256, 256, 0, stream>>>(XW, isd, out);

    const int scat_threads = N_EDGES * 32;   // 25.6M threads
    GCN_scatter_kernel<<<(scat_threads + 255) / 256, 256, 0, stream>>>(esrc, edst, XW, isd, out);
}